// HierarchicalMemorySystem_81003083202867
// MI455X (gfx1250) — compile-verified
//
#include <hip/hip_runtime.h>

typedef __attribute__((ext_vector_type(2))) float v2f;
typedef __attribute__((ext_vector_type(8))) float v8f;
typedef __attribute__((ext_vector_type(4))) int   v4i;

#define B_TOT   1024
#define M_TOT   131072
#define D_DIM   64
#define K_SEL   32
#define CHUNK   512
#define NCHUNK  (M_TOT / CHUNK)          // 256
#define CAND_PR (NCHUNK * K_SEL)         // 8192 candidates per row
#define SSTR    516                      // padded score row stride (dwords)
#define BSTR    68                       // padded B-stage row stride (dwords) = 256B + 16B
#define NSTRIP  (CHUNK / 128)            // 4 strips of 128 cols

// ---- CDNA5 async global->LDS copy (ASYNCcnt path), builtin or inline asm ----
__device__ __forceinline__ void async_cp16(const float* g, float* l) {
#if __has_builtin(__builtin_amdgcn_global_load_async_to_lds_b128)
  typedef __attribute__((address_space(1))) v4i* gp_t;   // global src
  typedef __attribute__((address_space(3))) v4i* lp_t;   // LDS dst
  __builtin_amdgcn_global_load_async_to_lds_b128(
      (gp_t)(v4i*)const_cast<float*>(g), (lp_t)(v4i*)l, 0, 0);
#else
  unsigned loff = (unsigned)(unsigned long long)(__attribute__((address_space(3))) float*)l;
  asm volatile("global_load_async_to_lds_b128 %0, %1, off"
               :: "v"(loff), "v"(g) : "memory");
#endif
}

template <int N>
__device__ __forceinline__ void wait_async() {
#if __has_builtin(__builtin_amdgcn_s_wait_asynccnt)
  __builtin_amdgcn_s_wait_asynccnt(N);
#else
  asm volatile("s_wait_asynccnt %0" :: "i"(N) : "memory");
#endif
}

// ---------------------------------------------------------------------------
// Kernel 1: query_proj = tanh(query @ Wk^T + bk)   [1024 x 64]
// ---------------------------------------------------------------------------
__global__ __launch_bounds__(128)
void proj_kernel(const float* __restrict__ q, const float* __restrict__ Wk,
                 const float* __restrict__ bk, float* __restrict__ proj) {
  const int lane = threadIdx.x & 31;
  const int wave = threadIdx.x >> 5;     // 0..3 -> output col tile
  const int b0   = blockIdx.x * 16;
  const int n0   = wave * 16;
  const int m    = lane & 15;
  const int hi   = lane >> 4;

  const v2f* arow = (const v2f*)(q  + (size_t)(b0 + m) * D_DIM);
  const v2f* brow = (const v2f*)(Wk + (size_t)(n0 + m) * D_DIM); // B(k,n)=Wk[n][k]

  v8f c = {};
#pragma unroll
  for (int kk = 0; kk < 16; ++kk) {
    v2f a = arow[2 * kk + hi];
    v2f b = brow[2 * kk + hi];
    c = __builtin_amdgcn_wmma_f32_16x16x4_f32(false, a, false, b,
                                              (short)0, c, false, false);
  }
  const float bias = bk[n0 + m];
#pragma unroll
  for (int r = 0; r < 8; ++r) {
    int row = b0 + r + 8 * hi;
    proj[(size_t)row * D_DIM + n0 + m] = tanhf(c[r] + bias);
  }
}

// ---------------------------------------------------------------------------
// Kernel 2: fused scores + per-chunk exact top-32, async-LDS staged B tiles.
// ---------------------------------------------------------------------------
__global__ __launch_bounds__(256)
void score_select_kernel(const float* __restrict__ proj,
                         const float* __restrict__ mem,
                         const float* __restrict__ ts,
                         float* __restrict__ cand_val,
                         int* __restrict__ cand_idx) {
  __shared__ float sc[16 * SSTR];             // 33 KB score tile (padded)
  __shared__ float tw[CHUNK];                 // 2 KB recency weights
  __shared__ float stage[2][128 * BSTR];      // 2 x 34 KB double-buffered B strips

  const int tid  = threadIdx.x;
  const int lane = tid & 31;
  const int wave = tid >> 5;                  // 0..7
  const int c0   = blockIdx.x * CHUNK;
  const int b0   = blockIdx.y * 16;
  const int m    = lane & 15;
  const int hi   = lane >> 4;

  // kick off strip 0 staging immediately (8 x b128 per wave, ASYNCcnt += 8)
  {
#pragma unroll
    for (int i = 0; i < 8; ++i) {
      int flat = i * 256 + tid;               // 16B segment id within strip
      int row  = flat >> 4;                   // 0..127
      int seg  = flat & 15;                   // 0..15 (16B units in a 256B row)
      async_cp16(mem + (size_t)(c0 + row) * D_DIM + seg * 4,
                 &stage[0][row * BSTR + seg * 4]);
    }
  }

  // recency weights: sigmoid(-t * 0.01)
  for (int i = tid; i < CHUNK; i += 256) {
    float t = ts[c0 + i];
    tw[i] = 1.0f / (1.0f + __expf(t * 0.01f));
  }

  // cache A fragments (query_proj rows b0..b0+15) for all 16 K-steps
  v2f afrag[16];
  {
    const v2f* arow = (const v2f*)(proj + (size_t)(b0 + m) * D_DIM);
#pragma unroll
    for (int kk = 0; kk < 16; ++kk) afrag[kk] = arow[2 * kk + hi];
  }

  for (int s = 0; s < NSTRIP; ++s) {
    if (s + 1 < NSTRIP) {                     // prefetch next strip into other buffer
#pragma unroll
      for (int i = 0; i < 8; ++i) {
        int flat = i * 256 + tid;
        int row  = flat >> 4;
        int seg  = flat & 15;
        async_cp16(mem + (size_t)(c0 + (s + 1) * 128 + row) * D_DIM + seg * 4,
                   &stage[(s + 1) & 1][row * BSTR + seg * 4]);
      }
      wait_async<8>();                        // strip s (first 8 ops) complete
    } else {
      wait_async<0>();
    }
    __syncthreads();                          // all waves' strip-s data visible

    const v2f* bls = (const v2f*)&stage[s & 1][(wave * 16 + m) * BSTR];
    v8f c = {};
#pragma unroll
    for (int kk = 0; kk < 16; ++kk) {
      v2f b = bls[2 * kk + hi];               // ds_load_b64, conflict-free
      c = __builtin_amdgcn_wmma_f32_16x16x4_f32(false, afrag[kk], false, b,
                                                (short)0, c, false, false);
    }
    const int colL = s * 128 + wave * 16 + m;
    const float w = tw[colL];
#pragma unroll
    for (int r = 0; r < 8; ++r) {
      int rowL = r + 8 * hi;
      sc[rowL * SSTR + colL] = 0.5f * c[r] + 0.5f * w;
    }
    __syncthreads();                          // strip s consumed; buffer reusable
  }

  // each wave extracts exact top-32 of two rows (512 values each)
  for (int rr = 0; rr < 2; ++rr) {
    const int row = wave * 2 + rr;
    float v[16];
#pragma unroll
    for (int j = 0; j < 16; ++j) v[j] = sc[row * SSTR + lane + 32 * j];

    float keep_v = -__builtin_inff();
    int   keep_i = 0;
    for (int it = 0; it < K_SEL; ++it) {
      float bv = -__builtin_inff(); int bs = 0;
#pragma unroll
      for (int j = 0; j < 16; ++j) if (v[j] > bv) { bv = v[j]; bs = j; }
      const int bidx = c0 + lane + 32 * bs;   // unique global slot id

      float rv = bv; int ridx = bidx;
#pragma unroll
      for (int off = 16; off > 0; off >>= 1) {
        float ov = __shfl_xor(rv, off, 32);
        int   oi = __shfl_xor(ridx, off, 32);
        if (ov > rv || (ov == rv && oi < ridx)) { rv = ov; ridx = oi; }
      }
      if (ridx == bidx) {                     // winner lane retires its slot
#pragma unroll
        for (int j = 0; j < 16; ++j) if (j == bs) v[j] = -__builtin_inff();
      }
      if (lane == it) { keep_v = rv; keep_i = ridx; }
    }
    const size_t base = ((size_t)(b0 + row) * NCHUNK + blockIdx.x) * K_SEL;
    cand_val[base + lane] = keep_v;           // descending order
    cand_idx[base + lane] = keep_i;
  }
}

// ---------------------------------------------------------------------------
// Kernel 3: per-row reduce 8192 candidates -> top-32, then gather memory rows.
// ---------------------------------------------------------------------------
__global__ __launch_bounds__(256)
void final_topk_gather_kernel(const float* __restrict__ cand_val,
                              const int* __restrict__ cand_idx,
                              const float* __restrict__ mem,
                              float* __restrict__ out) {
  __shared__ float wv[8 * 32];
  __shared__ int   wi[8 * 32];
  __shared__ int   sel[32];

  const int b    = blockIdx.x;
  const int tid  = threadIdx.x;
  const int lane = tid & 31;
  const int wave = tid >> 5;

  // stage 1: each wave reduces its 1024 candidates to top-32
  {
    const size_t base = (size_t)b * CAND_PR + (size_t)wave * 1024;
    float v[32]; int gi[32];
#pragma unroll
    for (int j = 0; j < 32; ++j) {
      v[j]  = cand_val[base + lane + 32 * j];
      gi[j] = cand_idx[base + lane + 32 * j];
    }
    float keep_v = -__builtin_inff(); int keep_i = 0;
    for (int it = 0; it < 32; ++it) {
      float bv = -__builtin_inff(); int bi = 0; int bs = 0;
#pragma unroll
      for (int j = 0; j < 32; ++j) if (v[j] > bv) { bv = v[j]; bi = gi[j]; bs = j; }
      float rv = bv; int ri = bi; int rl = lane;
#pragma unroll
      for (int off = 16; off > 0; off >>= 1) {
        float ov = __shfl_xor(rv, off, 32);
        int   oi = __shfl_xor(ri, off, 32);
        int   ol = __shfl_xor(rl, off, 32);
        if (ov > rv || (ov == rv && oi < ri)) { rv = ov; ri = oi; rl = ol; }
      }
      if (lane == rl) {
#pragma unroll
        for (int j = 0; j < 32; ++j) if (j == bs) v[j] = -__builtin_inff();
      }
      if (lane == it) { keep_v = rv; keep_i = ri; }
    }
    wv[wave * 32 + lane] = keep_v;
    wi[wave * 32 + lane] = keep_i;
  }
  __syncthreads();

  // stage 2: wave 0 reduces 256 -> final top-32 indices
  if (wave == 0) {
    float v[8]; int gi[8];
#pragma unroll
    for (int j = 0; j < 8; ++j) {
      v[j]  = wv[lane + 32 * j];
      gi[j] = wi[lane + 32 * j];
    }
    int keep_i = 0;
    for (int it = 0; it < 32; ++it) {
      float bv = -__builtin_inff(); int bi = 0; int bs = 0;
#pragma unroll
      for (int j = 0; j < 8; ++j) if (v[j] > bv) { bv = v[j]; bi = gi[j]; bs = j; }
      float rv = bv; int ri = bi; int rl = lane;
#pragma unroll
      for (int off = 16; off > 0; off >>= 1) {
        float ov = __shfl_xor(rv, off, 32);
        int   oi = __shfl_xor(ri, off, 32);
        int   ol = __shfl_xor(rl, off, 32);
        if (ov > rv || (ov == rv && oi < ri)) { rv = ov; ri = oi; rl = ol; }
      }
      if (lane == rl) {
#pragma unroll
        for (int j = 0; j < 8; ++j) if (j == bs) v[j] = -__builtin_inff();
      }
      if (lane == it) keep_i = ri;
    }
    sel[lane] = keep_i;
  }
  __syncthreads();

  // gather: out[b, i, :] = memory[sel[i], :]  (256 threads x 32B, coalesced)
  const int i   = tid >> 3;
  const int off = (tid & 7) * 8;
  const int s   = sel[i];
  const float4* src = (const float4*)(mem + (size_t)s * D_DIM + off);
  float4* dst = (float4*)(out + ((size_t)b * K_SEL + i) * D_DIM + off);
  dst[0] = src[0];
  dst[1] = src[1];
}

// ---------------------------------------------------------------------------
extern "C" void kernel_launch(void* const* d_in, const int* in_sizes, int n_in,
                              void* d_out, int out_size, void* d_ws, size_t ws_size,
                              hipStream_t stream) {
  (void)in_sizes; (void)n_in; (void)out_size; (void)ws_size;
  const float* query  = (const float*)d_in[0];   // [1024, 64]
  const float* memory = (const float*)d_in[1];   // [131072, 64]
  const float* ts     = (const float*)d_in[2];   // [131072]
  const float* Wk     = (const float*)d_in[3];   // [64, 64]
  const float* bk     = (const float*)d_in[4];   // [64]
  // d_in[5] = k (always 32 per setup)
  float* out = (float*)d_out;                    // [1024, 32, 64]

  float* proj     = (float*)d_ws;                               // 256 KB
  float* cand_val = proj + (size_t)B_TOT * D_DIM;               // 32 MB
  int*   cand_idx = (int*)(cand_val + (size_t)B_TOT * CAND_PR); // 32 MB

  proj_kernel<<<B_TOT / 16, 128, 0, stream>>>(query, Wk, bk, proj);

  dim3 g2(NCHUNK, B_TOT / 16);
  score_select_kernel<<<g2, 256, 0, stream>>>(proj, memory, ts, cand_val, cand_idx);

  final_topk_gather_kernel<<<B_TOT, 256, 0, stream>>>(cand_val, cand_idx, memory, out);
}